// CFKD_33285996544551
// MI455X (gfx1250) — compile-verified
//
#include <hip/hip_runtime.h>
#include <hip/hip_bf16.h>

// Problem constants from the reference
#define BATCH   512
#define OUT_DIM 512
#define IN_DIM  1024

typedef __attribute__((ext_vector_type(2))) float v2f;
typedef __attribute__((ext_vector_type(8))) float v8f;

// D(16x16,f32) = A(16x4,f32) x B(4x16,f32) + C  -- full-precision matrix pipe
__device__ __forceinline__ v8f wmma_f32(v2f a, v2f b, v8f c) {
  return __builtin_amdgcn_wmma_f32_16x16x4_f32(
      /*neg_a=*/false, a, /*neg_b=*/false, b,
      /*c_mod=*/(short)0, c, /*reuse_a=*/false, /*reuse_b=*/false);
}

__device__ __forceinline__ float hard_mask(float m) {
  // round(clip(m,0,1)); rintf = round-half-even, matches jnp.round
  return rintf(fminf(fmaxf(m, 0.0f), 1.0f));
}

// ---------------------------------------------------------------------------
// Kernel 1: C = U^T X / BATCH  (GEMM over b), fused epilogue:
//   Bm[k,d] = mask(M[k,d]);  Bw[k,d] = -2 * mask * C[k,d]
// Tiles: M-dim = k (OUT_DIM), N-dim = d (IN_DIM). Each wave: 16k x 32d.
// Waves: 32 ktiles * 32 dgroups = 1024  -> 128 blocks x 256 threads.
// ---------------------------------------------------------------------------
__global__ void k1_centroid_bhat(const float* __restrict__ X,
                                 const float* __restrict__ U,
                                 const float* __restrict__ M,
                                 float* __restrict__ Bm,
                                 float* __restrict__ Bw) {
  const int lane = threadIdx.x & 31;
  const int wid  = (blockIdx.x * blockDim.x + threadIdx.x) >> 5;
  const int half = lane >> 4;        // 0: K=0,1 ; 1: K=2,3 (32-bit A/B layout)
  const int lr   = lane & 15;

  const int kbase = (wid >> 5) * 16;   // 32 k-tiles
  const int dbase = (wid & 31) * 32;   // 32 d-groups of 2 tiles

  v8f acc0 = {};
  v8f acc1 = {};

  for (int b4 = 0; b4 < BATCH; b4 += 4) {
    const int r0 = b4 + half * 2;
    const int r1 = r0 + 1;
    // A[m=lr][kk=half*2+{0,1}] = U[b=r][k=kbase+lr]   (coalesced 64B per half)
    v2f a;
    a.x = U[r0 * OUT_DIM + kbase + lr];
    a.y = U[r1 * OUT_DIM + kbase + lr];
    // B[kk][n=lr] = X[b=r][d]
    v2f b0, b1;
    b0.x = X[r0 * IN_DIM + dbase + lr];
    b0.y = X[r1 * IN_DIM + dbase + lr];
    b1.x = X[r0 * IN_DIM + dbase + 16 + lr];
    b1.y = X[r1 * IN_DIM + dbase + 16 + lr];
    acc0 = wmma_f32(a, b0, acc0);
    acc1 = wmma_f32(a, b1, acc1);
  }

  const float inv = 1.0f / (float)BATCH;
#pragma unroll
  for (int v = 0; v < 8; ++v) {
    const int k = kbase + v + half * 8;      // D row = m
    {
      const int d = dbase + lr;              // D col = n
      const float c = acc0[v] * inv;
      const float m = hard_mask(M[k * IN_DIM + d]);
      Bm[k * IN_DIM + d] = m;
      Bw[k * IN_DIM + d] = -2.0f * m * c;
    }
    {
      const int d = dbase + 16 + lr;
      const float c = acc1[v] * inv;
      const float m = hard_mask(M[k * IN_DIM + d]);
      Bm[k * IN_DIM + d] = m;
      Bw[k * IN_DIM + d] = -2.0f * m * c;
    }
  }
}

// ---------------------------------------------------------------------------
// Kernel 2: Cm2[k] = 0.25 * sum_d Bw[k,d]^2   (= sum_d mask*C^2)
// ---------------------------------------------------------------------------
__global__ void k2_cm2(const float* __restrict__ Bw, float* __restrict__ Cm2) {
  __shared__ float red[256];
  const int k = blockIdx.x;
  float s = 0.0f;
  for (int d = threadIdx.x; d < IN_DIM; d += 256) {
    const float w = Bw[k * IN_DIM + d];
    s += w * w;
  }
  red[threadIdx.x] = s;
  __syncthreads();
  for (int off = 128; off > 0; off >>= 1) {
    if ((int)threadIdx.x < off) red[threadIdx.x] += red[threadIdx.x + off];
    __syncthreads();
  }
  if (threadIdx.x == 0) Cm2[k] = 0.25f * red[0];
}

// ---------------------------------------------------------------------------
// Kernel 3: D[b,k] = sum_d mask*X^2  +  sum_d X*Bw  +  Cm2[k]
// Single logical GEMM with Khat=2048 split into two K-loops sharing the
// accumulators. M-dim = b, N-dim = k. Each wave: 16b x 32k.
// Waves: 32 btiles * 16 kgroups = 512 -> 64 blocks x 256 threads.
// A and B per-lane K-pairs are contiguous -> float2 (global_load_b64) loads.
// ---------------------------------------------------------------------------
__global__ void k3_dist(const float* __restrict__ X,
                        const float* __restrict__ Bm,
                        const float* __restrict__ Bw,
                        const float* __restrict__ Cm2,
                        float* __restrict__ D) {
  const int lane = threadIdx.x & 31;
  const int wid  = (blockIdx.x * blockDim.x + threadIdx.x) >> 5;
  const int half = lane >> 4;
  const int lr   = lane & 15;

  const int bbase = (wid >> 4) * 16;   // 32 b-tiles
  const int kbase = (wid & 15) * 32;   // 16 k-groups of 2 tiles

  const float* xrow  = X  + (size_t)(bbase + lr) * IN_DIM;
  const float* bmr0  = Bm + (size_t)(kbase + lr) * IN_DIM;
  const float* bmr1  = Bm + (size_t)(kbase + 16 + lr) * IN_DIM;
  const float* bwr0  = Bw + (size_t)(kbase + lr) * IN_DIM;
  const float* bwr1  = Bw + (size_t)(kbase + 16 + lr) * IN_DIM;

  v8f acc0 = {};
  v8f acc1 = {};

  // Pass 1: acc += (X^2) * mask^T
#pragma unroll 4
  for (int d4 = 0; d4 < IN_DIM; d4 += 4) {
    const int dd = d4 + half * 2;
    v2f a = *(const v2f*)(xrow + dd);
    a.x *= a.x;
    a.y *= a.y;
    v2f b0 = *(const v2f*)(bmr0 + dd);
    v2f b1 = *(const v2f*)(bmr1 + dd);
    acc0 = wmma_f32(a, b0, acc0);
    acc1 = wmma_f32(a, b1, acc1);
  }

  // Pass 2: acc += X * (-2*mask*C)^T
#pragma unroll 4
  for (int d4 = 0; d4 < IN_DIM; d4 += 4) {
    const int dd = d4 + half * 2;
    v2f a  = *(const v2f*)(xrow + dd);
    v2f b0 = *(const v2f*)(bwr0 + dd);
    v2f b1 = *(const v2f*)(bwr1 + dd);
    acc0 = wmma_f32(a, b0, acc0);
    acc1 = wmma_f32(a, b1, acc1);
  }

  const float c0 = Cm2[kbase + lr];        // D col n = lane -> one scalar/lane
  const float c1 = Cm2[kbase + 16 + lr];
#pragma unroll
  for (int v = 0; v < 8; ++v) {
    const int brow = bbase + v + half * 8;
    D[brow * OUT_DIM + kbase + lr]      = acc0[v] + c0;
    D[brow * OUT_DIM + kbase + 16 + lr] = acc1[v] + c1;
  }
}

// ---------------------------------------------------------------------------
extern "C" void kernel_launch(void* const* d_in, const int* in_sizes, int n_in,
                              void* d_out, int out_size, void* d_ws, size_t ws_size,
                              hipStream_t stream) {
  (void)in_sizes; (void)n_in; (void)out_size; (void)ws_size;
  const float* X = (const float*)d_in[0];   // [512, 1024]
  const float* U = (const float*)d_in[1];   // [512, 512]
  const float* M = (const float*)d_in[2];   // [512, 1024]
  float* Dout = (float*)d_out;              // [512, 512]

  // Workspace layout: Bm (2 MB) | Bw (2 MB) | Cm2 (2 KB)
  float* Bm  = (float*)d_ws;
  float* Bw  = Bm + (size_t)OUT_DIM * IN_DIM;
  float* Cm2 = Bw + (size_t)OUT_DIM * IN_DIM;

  // K1: 1024 waves (32 ktiles x 32 dgroups) -> 128 blocks x 256
  k1_centroid_bhat<<<128, 256, 0, stream>>>(X, U, M, Bm, Bw);
  // K2: one block per k
  k2_cm2<<<OUT_DIM, 256, 0, stream>>>(Bw, Cm2);
  // K3: 512 waves (32 btiles x 16 kgroups) -> 64 blocks x 256
  k3_dist<<<64, 256, 0, stream>>>(X, Bm, Bw, Cm2, Dout);
}